// D_point_AE_fc_conv_33706903339268
// MI455X (gfx1250) — compile-verified
//
#include <hip/hip_runtime.h>

// ---------------------------------------------------------------------------
// PointNet++ AE forward for MI455X (gfx1250), fp32 end-to-end.
// Matrix ops: V_WMMA_F32_16X16X4_F32, wave32. Activations live in a
// K-interleaved layout (C4/4, M, 4) so each lane's B operand is one aligned
// b64 load; weights are pre-packed zero-padded (O, C4) so the K-loop has no
// guards/tail. Inner loop: 5 b64 loads (double-buffered) + 4 independent
// WMMAs; epilogue: 2 b128 stores per accumulator.
// ---------------------------------------------------------------------------

#define BB 16          // batch
#define NPTS 1024      // input points
#define EPSBN 1e-5f

typedef float v2f __attribute__((ext_vector_type(2)));
typedef float v4f __attribute__((ext_vector_type(4)));
typedef float v8f __attribute__((ext_vector_type(8)));

#if __has_builtin(__builtin_amdgcn_wmma_f32_16x16x4_f32)
#define HAVE_WMMA_F32_16X16X4 1
#endif

#if defined(HAVE_WMMA_F32_16X16X4) && defined(__HIP_DEVICE_COMPILE__)
#define USE_WMMA 1
#define WMMA_F32(A, Bv, Cacc)                                                  \
    __builtin_amdgcn_wmma_f32_16x16x4_f32(false, (A), false, (Bv), (short)0,   \
                                          (Cacc), false, false)
#endif

// ---------------------------------------------------------------------------
// Conv GEMM (1x1 conv): Y[o,m] = sum_c Wp[o,c] * X(c,m) + bias[o]
//  Wp : (O, C4) row-major, zero-padded (C4 % 4 == 0)
//  X  : per batch, K-interleaved (C4/4, M, 4): (c,m) -> ((c>>2)*M + m)*4 + (c&3)
//  Y  : per batch, K-interleaved (O/4,  M, 4)
// blockDim (32,4); grid (ceil(M/64), O/64, nBatch); M%16==0, O%64==0.
// ---------------------------------------------------------------------------
__device__ __forceinline__ void store_tile_conv(float* __restrict__ Yb,
                                                const v8f& acc,
                                                const float* __restrict__ bias,
                                                int ob, long M, long ybase)
{
    v4f lo, hi;
#pragma unroll
    for (int r = 0; r < 4; ++r) {
        lo[r] = acc[r]     + (bias ? bias[ob + r]     : 0.f);
        hi[r] = acc[r + 4] + (bias ? bias[ob + 4 + r] : 0.f);
    }
    *(v4f*)(Yb + (long)(ob >> 2) * M * 4 + ybase)       = lo;   // ob % 8 == 0
    *(v4f*)(Yb + ((long)(ob >> 2) + 1) * M * 4 + ybase) = hi;
}

__global__ void gemm_wmma_conv(const float* __restrict__ Wp,
                               const float* __restrict__ bias,
                               const float* __restrict__ X,
                               float* __restrict__ Y,
                               int O, int C4, int M,
                               long xBatch, long yBatch)
{
    const int lane = threadIdx.x;                       // 0..31
    const int m0   = (blockIdx.x * blockDim.y + threadIdx.y) * 16;
    if (m0 >= M) return;                                // wave-uniform
    const int o0   = blockIdx.y * 64;
    const int b    = blockIdx.z;

    const float* __restrict__ Xb = X + (long)b * xBatch;
    float* __restrict__       Yb = Y + (long)b * yBatch;

    const int  half = lane >> 4;      // 0: lanes 0-15, 1: lanes 16-31
    const int  l16  = lane & 15;
    const long mcol = m0 + l16;
    const int  ka0  = 2 * half;       // lane's K offset within a 4-step

    const float* wq0 = Wp + (long)(o0      + l16) * C4 + ka0;
    const float* wq1 = Wp + (long)(o0 + 16 + l16) * C4 + ka0;
    const float* wq2 = Wp + (long)(o0 + 32 + l16) * C4 + ka0;
    const float* wq3 = Wp + (long)(o0 + 48 + l16) * C4 + ka0;
    const float* xq  = Xb + mcol * 4 + ka0;   // K-group 0
    const long   xstep = (long)M * 4;         // advance one K-group

    v8f acc0 = {0.f,0.f,0.f,0.f,0.f,0.f,0.f,0.f};
    v8f acc1 = acc0, acc2 = acc0, acc3 = acc0;

#if defined(USE_WMMA)
    v2f a0 = *(const v2f*)wq0;
    v2f a1 = *(const v2f*)wq1;
    v2f a2 = *(const v2f*)wq2;
    v2f a3 = *(const v2f*)wq3;
    v2f bv = *(const v2f*)xq;
    for (int c0 = 4; c0 < C4; c0 += 4) {
        wq0 += 4; wq1 += 4; wq2 += 4; wq3 += 4; xq += xstep;
        const v2f na0 = *(const v2f*)wq0;     // next step's operands: issued
        const v2f na1 = *(const v2f*)wq1;     // before current WMMAs consume
        const v2f na2 = *(const v2f*)wq2;     // the previous loads
        const v2f na3 = *(const v2f*)wq3;
        const v2f nbv = *(const v2f*)xq;
        acc0 = WMMA_F32(a0, bv, acc0);
        acc1 = WMMA_F32(a1, bv, acc1);
        acc2 = WMMA_F32(a2, bv, acc2);
        acc3 = WMMA_F32(a3, bv, acc3);
        a0 = na0; a1 = na1; a2 = na2; a3 = na3; bv = nbv;
    }
    acc0 = WMMA_F32(a0, bv, acc0);
    acc1 = WMMA_F32(a1, bv, acc1);
    acc2 = WMMA_F32(a2, bv, acc2);
    acc3 = WMMA_F32(a3, bv, acc3);
#else
    for (int c = 0; c < C4; ++c) {
        const float xv = Xb[((long)(c >> 2) * M + mcol) * 4 + (c & 3)];
        for (int r = 0; r < 8; ++r) {
            const int oloc = half * 8 + r;
            acc0[r] += Wp[(long)(o0      + oloc) * C4 + c] * xv;
            acc1[r] += Wp[(long)(o0 + 16 + oloc) * C4 + c] * xv;
            acc2[r] += Wp[(long)(o0 + 32 + oloc) * C4 + c] * xv;
            acc3[r] += Wp[(long)(o0 + 48 + oloc) * C4 + c] * xv;
        }
    }
#endif

    const long ybase = mcol * 4;
    store_tile_conv(Yb, acc0, bias, o0      + half * 8, M, ybase);
    store_tile_conv(Yb, acc1, bias, o0 + 16 + half * 8, M, ybase);
    store_tile_conv(Yb, acc2, bias, o0 + 32 + half * 8, M, ybase);
    store_tile_conv(Yb, acc3, bias, o0 + 48 + half * 8, M, ybase);
}

// ---------------------------------------------------------------------------
// Row-major GEMM (decoder FC): Y[m,o] = relu(sum_c X[m,c]*W[o,c] + bias[o])
//  X (M, C) row-major, Y (M, O) row-major, C % 4 == 0, O % 64 == 0.
// ---------------------------------------------------------------------------
__global__ void gemm_wmma_rm(const float* __restrict__ W,
                             const float* __restrict__ bias,
                             const float* __restrict__ X,
                             float* __restrict__ Y,
                             int O, int C, int M, int relu)
{
    const int lane = threadIdx.x;
    const int m0   = (blockIdx.x * blockDim.y + threadIdx.y) * 16;
    if (m0 >= M) return;
    const int o0   = blockIdx.y * 64;

    const int  half = lane >> 4;
    const int  l16  = lane & 15;
    const long mcol = m0 + l16;
    const int  ka0  = 2 * half;

    const float* wq0 = W + (long)(o0      + l16) * C + ka0;
    const float* wq1 = W + (long)(o0 + 16 + l16) * C + ka0;
    const float* wq2 = W + (long)(o0 + 32 + l16) * C + ka0;
    const float* wq3 = W + (long)(o0 + 48 + l16) * C + ka0;
    const float* xq  = X + mcol * C + ka0;

    v8f acc0 = {0.f,0.f,0.f,0.f,0.f,0.f,0.f,0.f};
    v8f acc1 = acc0, acc2 = acc0, acc3 = acc0;

#if defined(USE_WMMA)
    v2f a0 = *(const v2f*)wq0;
    v2f a1 = *(const v2f*)wq1;
    v2f a2 = *(const v2f*)wq2;
    v2f a3 = *(const v2f*)wq3;
    v2f bv = *(const v2f*)xq;
    for (int c0 = 4; c0 < C; c0 += 4) {
        wq0 += 4; wq1 += 4; wq2 += 4; wq3 += 4; xq += 4;
        const v2f na0 = *(const v2f*)wq0;
        const v2f na1 = *(const v2f*)wq1;
        const v2f na2 = *(const v2f*)wq2;
        const v2f na3 = *(const v2f*)wq3;
        const v2f nbv = *(const v2f*)xq;
        acc0 = WMMA_F32(a0, bv, acc0);
        acc1 = WMMA_F32(a1, bv, acc1);
        acc2 = WMMA_F32(a2, bv, acc2);
        acc3 = WMMA_F32(a3, bv, acc3);
        a0 = na0; a1 = na1; a2 = na2; a3 = na3; bv = nbv;
    }
    acc0 = WMMA_F32(a0, bv, acc0);
    acc1 = WMMA_F32(a1, bv, acc1);
    acc2 = WMMA_F32(a2, bv, acc2);
    acc3 = WMMA_F32(a3, bv, acc3);
#else
    for (int c = 0; c < C; ++c) {
        const float xv = X[mcol * C + c];
        for (int r = 0; r < 8; ++r) {
            const int oloc = half * 8 + r;
            acc0[r] += W[(long)(o0      + oloc) * C + c] * xv;
            acc1[r] += W[(long)(o0 + 16 + oloc) * C + c] * xv;
            acc2[r] += W[(long)(o0 + 32 + oloc) * C + c] * xv;
            acc3[r] += W[(long)(o0 + 48 + oloc) * C + c] * xv;
        }
    }
#endif

    float* yrow = Y + mcol * O;
#pragma unroll
    for (int t = 0; t < 4; ++t) {
        const v8f& acc = (t == 0) ? acc0 : (t == 1) ? acc1 : (t == 2) ? acc2 : acc3;
        const int ob = o0 + t * 16 + half * 8;
        v4f lo, hi;
#pragma unroll
        for (int r = 0; r < 4; ++r) {
            float v0 = acc[r]     + (bias ? bias[ob + r]     : 0.f);
            float v1 = acc[r + 4] + (bias ? bias[ob + 4 + r] : 0.f);
            if (relu) { v0 = fmaxf(v0, 0.f); v1 = fmaxf(v1, 0.f); }
            lo[r] = v0; hi[r] = v1;
        }
        *(v4f*)(yrow + ob)     = lo;      // (mcol*O + ob) % 4 == 0
        *(v4f*)(yrow + ob + 4) = hi;
    }
}

// ---------------------------------------------------------------------------
// Pack W (O,C) row-major -> Wp (O,C4) zero padded
// ---------------------------------------------------------------------------
__global__ void pack_w_kernel(const float* __restrict__ W,
                              float* __restrict__ Wp, int O, int C, int C4)
{
    const int t = blockIdx.x * blockDim.x + threadIdx.x;
    if (t >= O * C4) return;
    const int c = t % C4, o = t / C4;
    Wp[t] = (c < C) ? W[(long)o * C + c] : 0.f;
}

// ---------------------------------------------------------------------------
// (B,6,N) input -> packed xyz0 (B,N,3)
// ---------------------------------------------------------------------------
__global__ void extract_xyz_kernel(const float* __restrict__ in,
                                   float* __restrict__ xyz0,
                                   int N, int Cin, long total)
{
    long t = (long)blockIdx.x * blockDim.x + threadIdx.x;
    if (t >= total) return;
    int  c  = (int)(t % 3);
    long bn = t / 3;
    long n  = bn % N;
    long b  = bn / N;
    xyz0[t] = in[(b * Cin + c) * N + n];
}

// ---------------------------------------------------------------------------
// Farthest point sampling: one block per batch. dist cached in LDS.
// Matches jax scan: emits incoming 'far' each step (idx[0]=0).
// ---------------------------------------------------------------------------
__global__ void fps_kernel(const float* __restrict__ xyz, // (B,N,3)
                           int N, int npoint,
                           float* __restrict__ newXyz,    // (B,npoint,3)
                           int* __restrict__ outIdx)      // (B,npoint)
{
    const int b   = blockIdx.x;
    const int tid = threadIdx.x;
    const float* p = xyz + (size_t)b * N * 3;

    __shared__ float dist[NPTS];
    __shared__ float rv[256];
    __shared__ int   ri[256];
    __shared__ int   s_far;

    for (int n = tid; n < N; n += blockDim.x) dist[n] = 1e10f;
    if (tid == 0) s_far = 0;
    __syncthreads();

    for (int i = 0; i < npoint; ++i) {
        const int far = s_far;
        const float cx = p[far * 3 + 0];
        const float cy = p[far * 3 + 1];
        const float cz = p[far * 3 + 2];
        if (tid == 0) {
            outIdx[(size_t)b * npoint + i] = far;
            newXyz[((size_t)b * npoint + i) * 3 + 0] = cx;
            newXyz[((size_t)b * npoint + i) * 3 + 1] = cy;
            newXyz[((size_t)b * npoint + i) * 3 + 2] = cz;
        }
        float best = -1.f; int bestN = 0;
        for (int n = tid; n < N; n += blockDim.x) {
            const float dx = p[n * 3 + 0] - cx;
            const float dy = p[n * 3 + 1] - cy;
            const float dz = p[n * 3 + 2] - cz;
            float d = dx * dx + dy * dy + dz * dz;
            d = fminf(dist[n], d);
            dist[n] = d;
            if (d > best) { best = d; bestN = n; }
        }
        rv[tid] = best; ri[tid] = bestN;
        __syncthreads();
        for (int s = 128; s > 0; s >>= 1) {
            if (tid < s) {
                if (rv[tid + s] > rv[tid] ||
                    (rv[tid + s] == rv[tid] && ri[tid + s] < ri[tid])) {
                    rv[tid] = rv[tid + s]; ri[tid] = ri[tid + s];
                }
            }
            __syncthreads();
        }
        if (tid == 0) s_far = ri[0];
        __syncthreads();
    }
}

// ---------------------------------------------------------------------------
// Ball query: thread per (b,s). Ascending indices with d<=r^2, pad with first.
// ---------------------------------------------------------------------------
__global__ void ball_query_kernel(const float* __restrict__ xyz,    // (B,N,3)
                                  const float* __restrict__ newXyz, // (B,S,3)
                                  int N, int S, float r2, int K,
                                  int* __restrict__ out)            // (B,S,K)
{
    const int t = blockIdx.x * blockDim.x + threadIdx.x;
    if (t >= BB * S) return;
    const int b = t / S, s = t - b * S;
    const float* p = xyz + (size_t)b * N * 3;
    const float* q = newXyz + ((size_t)b * S + s) * 3;
    const float qx = q[0], qy = q[1], qz = q[2];
    int* o = out + ((size_t)b * S + s) * K;
    int cnt = 0, first = 0;
    for (int n = 0; n < N && cnt < K; ++n) {
        const float dx = p[n * 3 + 0] - qx;
        const float dy = p[n * 3 + 1] - qy;
        const float dz = p[n * 3 + 2] - qz;
        if (dx * dx + dy * dy + dz * dz <= r2) {
            if (cnt == 0) first = n;
            o[cnt++] = n;
        }
    }
    for (int j = cnt; j < K; ++j) o[j] = first;
}

// ---------------------------------------------------------------------------
// Build grouped tensor X, K-interleaved (B, C4/4, M=K*S, 4), zero padded:
//  c<3 : centered xyz gather ; 3<=c<C : feature gather (pts laid out (B,D,N))
// t is the OUTPUT flat index (coalesced stores).
// ---------------------------------------------------------------------------
__global__ void group_kernel(const float* __restrict__ xyz,     // (B,N,3)
                             const float* __restrict__ pts,     // base, (D,N)/batch
                             long ptsBatch,
                             const float* __restrict__ newXyz,  // (B,S,3)
                             const int* __restrict__ idx,       // (B,S,K)
                             int N, int S, int K, int C, int C4, long total,
                             float* __restrict__ Xout)
{
    long t = (long)blockIdx.x * blockDim.x + threadIdx.x;
    if (t >= total) return;
    const long M  = (long)K * S;
    const int  r  = (int)(t & 3);
    const long q  = t >> 2;
    const long m  = q % M;
    const long gg = q / M;
    const int  g  = (int)(gg % (C4 >> 2));
    const int  b  = (int)(gg / (C4 >> 2));
    const int  c  = g * 4 + r;
    const int  k  = (int)(m / S);
    const int  s  = (int)(m % S);
    float v = 0.f;
    if (c < C) {
        const int n = idx[((long)b * S + s) * K + k];
        if (c < 3)
            v = xyz[((long)b * N + n) * 3 + c] - newXyz[((long)b * S + s) * 3 + c];
        else
            v = pts[(long)b * ptsBatch + (long)(c - 3) * N + n];
    }
    Xout[t] = v;
}

// SA3 group_all: X interleaved (B, 260/4, 128, 4) = concat(new_xyz2^T, l2_pts)
__global__ void build_sa3_kernel(const float* __restrict__ nx2,   // (B,128,3)
                                 const float* __restrict__ l2pts, // (B,256,128)
                                 float* __restrict__ Xout, long total)
{
    long t = (long)blockIdx.x * blockDim.x + threadIdx.x;
    if (t >= total) return;
    const int  r  = (int)(t & 3);
    const long q  = t >> 2;
    const int  m  = (int)(q % 128);          // k index (S=1)
    const long gg = q / 128;
    const int  g  = (int)(gg % 65);          // 260/4
    const int  b  = (int)(gg / 65);
    const int  c  = g * 4 + r;
    float v = 0.f;
    if (c < 3)        v = nx2[((long)b * 128 + m) * 3 + c];
    else if (c < 259) v = l2pts[((long)b * 256 + (c - 3)) * 128 + m];
    Xout[t] = v;
}

// ---------------------------------------------------------------------------
// BatchNorm stats over (b,m) per channel o; Y interleaved (B, O/4, M, 4)
// ---------------------------------------------------------------------------
__global__ void bn_stats_kernel(const float* __restrict__ Y,
                                int O, long M, long nBM,
                                float* __restrict__ sums)    // [2*O]
{
    const int  o     = blockIdx.x;
    const int  tid   = threadIdx.x;
    const long obase = (long)(o >> 2) * M * 4 + (o & 3);
    const long bstr  = (long)O * M;
    float s = 0.f, ss = 0.f;
    for (long j = (long)blockIdx.y * blockDim.x + tid; j < nBM;
         j += (long)gridDim.y * blockDim.x) {
        const long b = j / M, m = j - b * M;
        const float v = Y[b * bstr + obase + m * 4];
        s += v; ss += v * v;
    }
    __shared__ float sh0[256], sh1[256];
    sh0[tid] = s; sh1[tid] = ss;
    __syncthreads();
    for (int st = 128; st > 0; st >>= 1) {
        if (tid < st) { sh0[tid] += sh0[tid + st]; sh1[tid] += sh1[tid + st]; }
        __syncthreads();
    }
    if (tid == 0) {
        atomicAdd(&sums[o],     sh0[0]);
        atomicAdd(&sums[O + o], sh1[0]);
    }
}

__global__ void bn_apply_relu_kernel(float* __restrict__ Y,
                                     const float* __restrict__ sums,
                                     const float* __restrict__ gamma,
                                     const float* __restrict__ beta,
                                     int O, long M, long total, float invCnt)
{
    long t = (long)blockIdx.x * blockDim.x + threadIdx.x;
    if (t >= total) return;
    const int r = (int)(t & 3);
    const int g = (int)(((t >> 2) / M) % (O >> 2));
    const int o = g * 4 + r;
    const float mean = sums[o] * invCnt;
    const float var  = sums[O + o] * invCnt - mean * mean;
    float v = gamma[o] * (Y[t] - mean) * rsqrtf(var + EPSBN) + beta[o];
    Y[t] = v > 0.f ? v : 0.f;
}

// max over K: Y interleaved (B, O/4, K*S, 4) -> out plain (B,O,S)
__global__ void maxk_kernel(const float* __restrict__ Y,
                            float* __restrict__ out,
                            int O, int K, int S, long total)
{
    long t = (long)blockIdx.x * blockDim.x + threadIdx.x;
    if (t >= total) return;
    const int s = (int)(t % S);
    const int o = (int)((t / S) % O);
    const int b = (int)(t / ((long)S * O));
    const float* p = Y + ((long)(b * (O >> 2) + (o >> 2)) * K * S) * 4 + (o & 3);
    float m = p[(long)s * 4];
    for (int k = 1; k < K; ++k) m = fmaxf(m, p[((long)k * S + s) * 4]);
    out[t] = m;
}

// l2_xyz output: d_out chunk (B,3,128) from nx2 (B,128,3)
__global__ void l2xyz_out_kernel(const float* __restrict__ nx2,
                                 float* __restrict__ out)
{
    const int t = blockIdx.x * blockDim.x + threadIdx.x;
    if (t >= BB * 3 * 128) return;
    const int s = t % 128;
    const int r = t / 128;
    const int c = r % 3;
    const int b = r / 3;
    out[t] = nx2[((long)b * 128 + s) * 3 + c];
}

__global__ void fill_kernel(float* __restrict__ p, long n, float v)
{
    long t = (long)blockIdx.x * blockDim.x + threadIdx.x;
    if (t < n) p[t] = v;
}

// ---------------------------------------------------------------------------
// Host orchestration
// ---------------------------------------------------------------------------
extern "C" void kernel_launch(void* const* d_in, const int* in_sizes, int n_in,
                              void* d_out, int out_size, void* d_ws, size_t ws_size,
                              hipStream_t stream)
{
    (void)in_sizes; (void)n_in; (void)out_size; (void)ws_size;

    // ---- inputs (setup_inputs() insertion order: xyz, then params
    //      sa1(3x[W,b,gamma,beta]), sa2, sa3, dec(3x[W,b])) ----
    const float* xyz_in = (const float*)d_in[0];
    const float *saW[9], *saB[9], *saG[9], *saBe[9];
    int p = 1;
    for (int i = 0; i < 9; ++i) {
        saW[i]  = (const float*)d_in[p++];
        saB[i]  = (const float*)d_in[p++];
        saG[i]  = (const float*)d_in[p++];
        saBe[i] = (const float*)d_in[p++];
    }
    const float *decW[3], *decB[3];
    for (int i = 0; i < 3; ++i) {
        decW[i] = (const float*)d_in[p++];
        decB[i] = (const float*)d_in[p++];
    }

    // per conv layer: O, C, C4
    const int convO[9]  = { 64,  64, 128, 128, 128, 256, 256,  256, 1024 };
    const int convC[9]  = {  6,  64,  64, 131, 128, 128, 259,  256,  256 };
    const int convC4[9] = {  8,  64,  64, 132, 128, 128, 260,  256,  256 };

    // ---- workspace bump allocator ----
    char* wp = (char*)d_ws;
    auto allocF = [&](size_t n) { float* q = (float*)wp; wp += n * sizeof(float); return q; };
    auto allocI = [&](size_t n) { int*   q = (int*)wp;   wp += n * sizeof(int);   return q; };

    float* bufA  = allocF(67108864);      // 16*128*64*512 (largest activation)
    float* bufB  = allocF(67108864);
    float* xyz0  = allocF((size_t)BB * 1024 * 3);
    float* nx1   = allocF((size_t)BB * 512 * 3);
    float* nx2   = allocF((size_t)BB * 128 * 3);
    float* l1pts = allocF((size_t)BB * 128 * 512);
    float* l2pts = allocF((size_t)BB * 256 * 128);
    float* l3pts = allocF((size_t)BB * 1024);
    float* dec1  = allocF((size_t)BB * 256);
    float* dec2  = allocF((size_t)BB * 256);
    float* stats = allocF(2048);
    float* wpk[9];
    for (int i = 0; i < 9; ++i) wpk[i] = allocF((size_t)convO[i] * convC4[i]);
    int* fidx1 = allocI((size_t)BB * 512);
    int* fidx2 = allocI((size_t)BB * 128);
    int* ball1 = allocI((size_t)BB * 512 * 64);
    int* ball2 = allocI((size_t)BB * 128 * 64);

    auto blocks = [](long total, int bs) { return (unsigned)((total + bs - 1) / bs); };

    // pack all conv weights (zero-padded K)
    for (int i = 0; i < 9; ++i) {
        const long tw = (long)convO[i] * convC4[i];
        pack_w_kernel<<<blocks(tw, 256), 256, 0, stream>>>(
            saW[i], wpk[i], convO[i], convC[i], convC4[i]);
    }

    auto gemmc = [&](int li, const float* X, float* Y, int M) {
        const int O = convO[li], C4 = convC4[li];
        dim3 blk(32, 4);
        dim3 grd((unsigned)((M + 63) / 64), (unsigned)(O / 64), BB);
        gemm_wmma_conv<<<grd, blk, 0, stream>>>(wpk[li], saB[li], X, Y,
                                                O, C4, M,
                                                (long)C4 * M, (long)O * M);
    };
    auto bn = [&](float* Y, int li, long M) {
        const int O = convO[li];
        fill_kernel<<<blocks(2 * O, 256), 256, 0, stream>>>(stats, 2 * O, 0.f);
        const long nBM = (long)BB * M;
        long ch = (nBM + 16383) / 16384; if (ch > 64) ch = 64; if (ch < 1) ch = 1;
        bn_stats_kernel<<<dim3((unsigned)O, (unsigned)ch), 256, 0, stream>>>(
            Y, O, M, nBM, stats);
        const long total = nBM * O;
        bn_apply_relu_kernel<<<blocks(total, 256), 256, 0, stream>>>(
            Y, stats, saG[li], saBe[li], O, M, total, 1.f / (float)nBM);
    };

    // ================= SA1 : N=1024 -> S=512, K=64, C=6/C4=8, M=32768 =======
    {
        const int N = 1024, S = 512, K = 64, C = 6, C4 = 8, M = K * S;
        extract_xyz_kernel<<<blocks((long)BB * N * 3, 256), 256, 0, stream>>>(
            xyz_in, xyz0, N, 6, (long)BB * N * 3);
        fps_kernel<<<BB, 256, 0, stream>>>(xyz0, N, S, nx1, fidx1);
        ball_query_kernel<<<blocks((long)BB * S, 256), 256, 0, stream>>>(
            xyz0, nx1, N, S, 0.015f * 0.015f, K, ball1);
        group_kernel<<<blocks((long)BB * C4 * M, 256), 256, 0, stream>>>(
            xyz0, xyz_in + 3 * N, (long)6 * N, nx1, ball1,
            N, S, K, C, C4, (long)BB * C4 * M, bufA);
        gemmc(0, bufA, bufB, M);  bn(bufB, 0, M);
        gemmc(1, bufB, bufA, M);  bn(bufA, 1, M);
        gemmc(2, bufA, bufB, M);  bn(bufB, 2, M);
        maxk_kernel<<<blocks((long)BB * 128 * S, 256), 256, 0, stream>>>(
            bufB, l1pts, 128, K, S, (long)BB * 128 * S);
    }

    // ================= SA2 : N=512 -> S=128, K=64, C=131/C4=132, M=8192 =====
    {
        const int N = 512, S = 128, K = 64, C = 131, C4 = 132, M = K * S;
        fps_kernel<<<BB, 256, 0, stream>>>(nx1, N, S, nx2, fidx2);
        ball_query_kernel<<<blocks((long)BB * S, 256), 256, 0, stream>>>(
            nx1, nx2, N, S, 0.04f * 0.04f, K, ball2);
        group_kernel<<<blocks((long)BB * C4 * M, 256), 256, 0, stream>>>(
            nx1, l1pts, (long)128 * N, nx2, ball2,
            N, S, K, C, C4, (long)BB * C4 * M, bufA);
        gemmc(3, bufA, bufB, M);  bn(bufB, 3, M);
        gemmc(4, bufB, bufA, M);  bn(bufA, 4, M);
        gemmc(5, bufA, bufB, M);  bn(bufB, 5, M);
        maxk_kernel<<<blocks((long)BB * 256 * S, 256), 256, 0, stream>>>(
            bufB, l2pts, 256, K, S, (long)BB * 256 * S);
    }

    // ================= SA3 : group_all, C=259/C4=260, K=128, S=1, M=128 =====
    {
        const int M = 128;
        build_sa3_kernel<<<blocks((long)BB * 260 * 128, 256), 256, 0, stream>>>(
            nx2, l2pts, bufA, (long)BB * 260 * 128);
        gemmc(6, bufA, bufB, M);  bn(bufB, 6, M);
        gemmc(7, bufB, bufA, M);  bn(bufA, 7, M);
        gemmc(8, bufA, bufB, M);  bn(bufB, 8, M);
        maxk_kernel<<<blocks((long)BB * 1024, 256), 256, 0, stream>>>(
            bufB, l3pts, 1024, 128, 1, (long)BB * 1024);
    }

    // ================= Decoder: (B,1024) -> 256 -> 256 -> 768 ===============
    float* outp = (float*)d_out;
    {
        dim3 blk(32, 4), grd1(1, 256 / 64, 1), grd3(1, 768 / 64, 1);
        gemm_wmma_rm<<<grd1, blk, 0, stream>>>(decW[0], decB[0], l3pts, dec1,
                                               256, 1024, BB, 1);
        gemm_wmma_rm<<<grd1, blk, 0, stream>>>(decW[1], decB[1], dec1, dec2,
                                               256, 256, BB, 1);
        gemm_wmma_rm<<<grd3, blk, 0, stream>>>(decW[2], decB[2], dec2, outp,
                                               768, 256, BB, 0);
    }

    // ================= remaining outputs ====================================
    l2xyz_out_kernel<<<blocks(BB * 3 * 128, 256), 256, 0, stream>>>(
        nx2, outp + BB * 768);
    fill_kernel<<<1, 64, 0, stream>>>(outp + BB * 768 + BB * 3 * 128,
                                      (long)BB * 3, 0.f);
}